// WarpedMSELoss_20186346291824
// MI455X (gfx1250) — compile-verified
//
#include <hip/hip_runtime.h>

// ---------------- problem constants ----------------
#define DVOL 192
constexpr int NVOX  = DVOL * DVOL * DVOL;   // 7,077,888
constexpr int BLOCK = 256;                  // 8 waves (wave32)
constexpr int VPT   = 4;                    // voxels per thread (along x)
constexpr int TILE  = BLOCK * VPT;          // 1024 voxels per tile
constexpr int NTILES = NVOX / TILE;         // 6912 (exact)
constexpr int TPB   = 4;                    // tiles per block (pipeline depth)
constexpr int GRID1 = NTILES / TPB;         // 1728 blocks
constexpr int ASYNC_PER_STAGE = 5;          // async instrs issued per wave per stage

#define AS_GLOBAL __attribute__((address_space(1)))
#define AS_LDS    __attribute__((address_space(3)))

typedef int v4i __attribute__((ext_vector_type(4)));

// ---------------- CDNA5 async-to-LDS + wait-counter plumbing ----------------
#if defined(__HIP_DEVICE_COMPILE__)
#  if __has_builtin(__builtin_amdgcn_global_load_async_to_lds_b128) && \
      __has_builtin(__builtin_amdgcn_global_load_async_to_lds_b32)
#    define HAVE_ASYNC_BUILTIN 1
#  else
#    warning "gfx1250 async-to-LDS builtins not found; using inline-asm fallback"
#  endif
#  if __has_builtin(__builtin_amdgcn_s_wait_asynccnt)
#    define WAIT_ASYNC(n) __builtin_amdgcn_s_wait_asynccnt(n)
#  else
#    define WAIT_ASYNC(n) asm volatile("s_wait_asynccnt %0" :: "n"(n) : "memory")
#  endif
#  if __has_builtin(__builtin_amdgcn_s_wait_dscnt)
#    define WAIT_DS(n) __builtin_amdgcn_s_wait_dscnt(n)
#  else
#    define WAIT_DS(n) asm volatile("s_wait_dscnt %0" :: "n"(n) : "memory")
#  endif
#else
#  define WAIT_ASYNC(n) ((void)0)
#  define WAIT_DS(n)    ((void)0)
#endif

__device__ __forceinline__ void async_b128(const void* g, void* l) {
#if !defined(__HIP_DEVICE_COMPILE__)
  (void)g; (void)l;
#elif defined(HAVE_ASYNC_BUILTIN)
  __builtin_amdgcn_global_load_async_to_lds_b128(
      (AS_GLOBAL v4i*)g, (AS_LDS v4i*)l, 0, 0);
#else
  unsigned lofs = (unsigned)(unsigned long long)(AS_LDS void*)l;
  asm volatile("global_load_async_to_lds_b128 %0, %1, off" :: "v"(lofs), "v"(g) : "memory");
#endif
}

__device__ __forceinline__ void async_b32(const void* g, void* l) {
#if !defined(__HIP_DEVICE_COMPILE__)
  (void)g; (void)l;
#elif defined(HAVE_ASYNC_BUILTIN)
  __builtin_amdgcn_global_load_async_to_lds_b32(
      (AS_GLOBAL int*)g, (AS_LDS int*)l, 0, 0);
#else
  unsigned lofs = (unsigned)(unsigned long long)(AS_LDS void*)l;
  asm volatile("global_load_async_to_lds_b32 %0, %1, off" :: "v"(lofs), "v"(g) : "memory");
#endif
}

// ---------------- kernel 1: warped-MSE per-block partials ----------------
__global__ __launch_bounds__(BLOCK) void warp_mse_partials(
    const float* __restrict__ mov,            // [192^3] moving image
    const float* __restrict__ vf,             // [3,192^3] displacement (z,y,x)
    const float* __restrict__ fixd,           // [192^3] fixed image
    const unsigned char* __restrict__ mask,   // [192^3] bool mask (1 byte)
    float* __restrict__ partials)             // [2*GRID1]: sums then counts
{
  // double-buffered streaming staging (2 x 17KB LDS)
  __shared__ alignas(16) float s_vz[2][TILE];
  __shared__ alignas(16) float s_vy[2][TILE];
  __shared__ alignas(16) float s_vx[2][TILE];
  __shared__ alignas(16) float s_fx[2][TILE];
  __shared__ alignas(16) unsigned int s_mk[2][BLOCK];

  const int tid = threadIdx.x;
  const int t0  = blockIdx.x * TPB;

  auto stage = [&](int t, int buf) {
    const int n0 = t * TILE + tid * VPT;            // 16B-aligned element base
    async_b128(vf            + n0, &s_vz[buf][tid * VPT]);
    async_b128(vf +     NVOX + n0, &s_vy[buf][tid * VPT]);
    async_b128(vf + 2 * NVOX + n0, &s_vx[buf][tid * VPT]);
    async_b128(fixd          + n0, &s_fx[buf][tid * VPT]);
    async_b32 (mask          + n0, &s_mk[buf][tid]);   // 4 packed mask bytes
  };

  stage(t0, 0);
  float sum = 0.f, cnt = 0.f;

  #pragma unroll
  for (int i = 0; i < TPB; ++i) {
    if (i + 1 < TPB) {
      WAIT_DS(0);                 // prior LDS reads done before async overwrites buffer
      stage(t0 + i + 1, (i + 1) & 1);
      WAIT_ASYNC(ASYNC_PER_STAGE);  // <=5 outstanding => stage i landed (in-order)
    } else {
      WAIT_ASYNC(0);
    }

    const int buf = i & 1;
    const int n0  = (t0 + i) * TILE + tid * VPT;
    const int xb  = n0 % DVOL;          // VPT chunk never crosses a row (192 % 4 == 0)
    const int rem = n0 / DVOL;
    const int yb  = rem % DVOL;
    const int zb  = rem / DVOL;
    const unsigned int mk = s_mk[buf][tid];

    #pragma unroll
    for (int v = 0; v < VPT; ++v) {
      const int sl = tid * VPT + v;
      const float fz = (float)zb       + s_vz[buf][sl];
      const float fy = (float)yb       + s_vy[buf][sl];
      const float fx = (float)(xb + v) + s_vx[buf][sl];

      const float z0f = floorf(fz), y0f = floorf(fy), x0f = floorf(fx);
      const float wz = fz - z0f, wy = fy - y0f, wx = fx - x0f;
      const int z0 = (int)z0f, y0 = (int)y0f, x0 = (int)x0f;

      // All 8 corners strictly inside the volume? (z0,y0,x0 in [0,190])
      const bool interior = ((unsigned)z0 < (unsigned)(DVOL - 1)) &&
                            ((unsigned)y0 < (unsigned)(DVOL - 1)) &&
                            ((unsigned)x0 < (unsigned)(DVOL - 1));

      float acc;
      if (__all(interior)) {
        // -------- wave-uniform fast path --------
        // one base address; the 8 gathers become immediate-offset loads;
        // nested lerp: 7 lerps instead of 8 weight products.
        const float* p = mov + ((z0 * DVOL + y0) * DVOL + x0);
        const float m000 = p[0];
        const float m001 = p[1];
        const float m010 = p[DVOL];
        const float m011 = p[DVOL + 1];
        const float m100 = p[DVOL * DVOL];
        const float m101 = p[DVOL * DVOL + 1];
        const float m110 = p[DVOL * DVOL + DVOL];
        const float m111 = p[DVOL * DVOL + DVOL + 1];
        const float a00 = fmaf(wx, m001 - m000, m000);
        const float a01 = fmaf(wx, m011 - m010, m010);
        const float a10 = fmaf(wx, m101 - m100, m100);
        const float a11 = fmaf(wx, m111 - m110, m110);
        const float a0  = fmaf(wy, a01 - a00, a00);
        const float a1  = fmaf(wy, a11 - a10, a10);
        acc = fmaf(wz, a1 - a0, a0);
      } else {
        // -------- general path: clamped addresses + per-axis validity
        //          folded into weights (zero padding semantics) --------
        int   cz[2], cy[2], cx[2];
        float wzv[2], wyv[2], wxv[2];
        cz[0] = min(max(z0, 0), DVOL - 1);     cz[1] = min(max(z0 + 1, 0), DVOL - 1);
        cy[0] = min(max(y0, 0), DVOL - 1);     cy[1] = min(max(y0 + 1, 0), DVOL - 1);
        cx[0] = min(max(x0, 0), DVOL - 1);     cx[1] = min(max(x0 + 1, 0), DVOL - 1);
        wzv[0] = ((unsigned)z0       < (unsigned)DVOL) ? (1.f - wz) : 0.f;
        wzv[1] = ((unsigned)(z0 + 1) < (unsigned)DVOL) ? wz         : 0.f;
        wyv[0] = ((unsigned)y0       < (unsigned)DVOL) ? (1.f - wy) : 0.f;
        wyv[1] = ((unsigned)(y0 + 1) < (unsigned)DVOL) ? wy         : 0.f;
        wxv[0] = ((unsigned)x0       < (unsigned)DVOL) ? (1.f - wx) : 0.f;
        wxv[1] = ((unsigned)(x0 + 1) < (unsigned)DVOL) ? wx         : 0.f;

        acc = 0.f;
        #pragma unroll
        for (int dz = 0; dz < 2; ++dz) {
          const int oz = cz[dz] * DVOL;
          #pragma unroll
          for (int dy = 0; dy < 2; ++dy) {
            const int   oy  = (oz + cy[dy]) * DVOL;
            const float wzy = wzv[dz] * wyv[dy];
            #pragma unroll
            for (int dx = 0; dx < 2; ++dx)
              acc = fmaf(mov[oy + cx[dx]], wzy * wxv[dx], acc);
          }
        }
      }

      const float m = ((mk >> (8 * v)) & 0xFFu) ? 1.f : 0.f;
      const float d = acc - s_fx[buf][sl];
      sum = fmaf(d * d, m, sum);
      cnt += m;
    }
  }

  // wave32 butterfly reduction, then cross-wave via LDS
  #pragma unroll
  for (int off = 16; off > 0; off >>= 1) {
    sum += __shfl_xor(sum, off, 32);
    cnt += __shfl_xor(cnt, off, 32);
  }
  __shared__ float red_s[BLOCK / 32], red_c[BLOCK / 32];
  const int wave = tid >> 5, lane = tid & 31;
  if (lane == 0) { red_s[wave] = sum; red_c[wave] = cnt; }
  __syncthreads();
  if (tid == 0) {
    float bs = 0.f, bc = 0.f;
    #pragma unroll
    for (int w = 0; w < BLOCK / 32; ++w) { bs += red_s[w]; bc += red_c[w]; }
    partials[blockIdx.x]         = bs;
    partials[GRID1 + blockIdx.x] = bc;
  }
}

// ---------------- kernel 2: deterministic final reduction ----------------
__global__ __launch_bounds__(BLOCK) void warp_mse_final(
    const float* __restrict__ partials, float* __restrict__ out)
{
  __shared__ double ssum[BLOCK], scnt[BLOCK];
  const int tid = threadIdx.x;
  double s = 0.0, c = 0.0;
  for (int i = tid; i < GRID1; i += BLOCK) {
    s += (double)partials[i];
    c += (double)partials[GRID1 + i];
  }
  ssum[tid] = s; scnt[tid] = c;
  __syncthreads();
  for (int off = BLOCK / 2; off > 0; off >>= 1) {
    if (tid < off) { ssum[tid] += ssum[tid + off]; scnt[tid] += scnt[tid + off]; }
    __syncthreads();
  }
  if (tid == 0) out[0] = (float)(ssum[0] / fmax(scnt[0], 1.0));
}

// ---------------- entry point ----------------
extern "C" void kernel_launch(void* const* d_in, const int* in_sizes, int n_in,
                              void* d_out, int out_size, void* d_ws, size_t ws_size,
                              hipStream_t stream) {
  (void)in_sizes; (void)n_in; (void)out_size; (void)ws_size;
  const float*         mov  = (const float*)d_in[0];          // moving_image  [192^3]
  const float*         vf   = (const float*)d_in[1];          // vector_field  [3,192^3]
  const float*         fixd = (const float*)d_in[2];          // fixed_image   [192^3]
  const unsigned char* mask = (const unsigned char*)d_in[3];  // bool mask     [192^3]
  float* partials = (float*)d_ws;                             // 2*GRID1 floats

  warp_mse_partials<<<GRID1, BLOCK, 0, stream>>>(mov, vf, fixd, mask, partials);
  warp_mse_final<<<1, BLOCK, 0, stream>>>(partials, (float*)d_out);
}